// HyperMLR_584115553038
// MI455X (gfx1250) — compile-verified
//
#include <hip/hip_runtime.h>
#include <math.h>

typedef float v2f __attribute__((ext_vector_type(2)));
typedef float v8f __attribute__((ext_vector_type(8)));

#define CCH   256      // channels (K of the GEMM)
#define HW    65536    // H*W
#define NCLS  19       // classes
#define KC    0.5f     // curvature

__global__ void __launch_bounds__(256, 2)
hyper_mlr_fused(const float* __restrict__ x,
                const float* __restrict__ P,
                const float* __restrict__ A,
                float* __restrict__ out)
{
    // 64 rows x 256 ch: rows 0-18 = -P, rows 32-50 = raw A.
    // Rows 19-31 / 51-63 feed only WMMA output rows for classes >= 19
    // (never stored), so their contents are don't-care -> overlay the
    // small per-class arrays there. Static LDS = exactly 64KB.
    __shared__ float W[64 * CCH];
    float* ppS  = &W[19 * CCH];        // pp[n] = ||P_n||^2
    float* pdS  = &W[19 * CCH + 32];   // sum(-P_n . A_n)/||A_n||
    float* anS  = &W[19 * CCH + 64];   // ||A_n||
    float* invS = &W[19 * CCH + 96];   // 1/max(||A_n||, eps)

    const int tid = threadIdx.x;

    // ---- stage -P and raw A into LDS ----
    for (int i = tid; i < NCLS * CCH; i += 256) {
        int n = i >> 8, c = i & 255;
        W[n * CCH + c]        = -P[i];
        W[(32 + n) * CCH + c] =  A[i];
    }
    __syncthreads();

    // ---- per-class reductions: 8 lanes/class, shuffle tree-combine ----
    if (tid < NCLS * 8) {
        int n    = tid >> 3;
        int part = tid & 7;
        const float* pr = &W[n * CCH + part * 32];
        const float* ar = &W[(32 + n) * CCH + part * 32];
        float sp = 0.f, sa = 0.f, spa = 0.f;
        for (int c = 0; c < 32; ++c) {
            float p = pr[c], a = ar[c];
            sp  = fmaf(p, p, sp);
            sa  = fmaf(a, a, sa);
            spa = fmaf(p, a, spa);
        }
        // groups of 8 are lane-aligned (152 = 19*8, waves fully cover groups)
        for (int d = 1; d < 8; d <<= 1) {
            sp  += __shfl_xor(sp,  d);
            sa  += __shfl_xor(sa,  d);
            spa += __shfl_xor(spa, d);
        }
        if (part == 0) {
            float an  = sqrtf(sa);
            float anc = fmaxf(an, 1e-12f);
            ppS[n]  = sp;
            anS[n]  = an;
            invS[n] = 1.0f / anc;
            pdS[n]  = spa / anc;          // alpha-term dot vs *normalized* A
        }
    }
    __syncthreads();

    // ---- fused dual-GEMM over this wave's 16-pixel tile ----
    const int lane = tid & 31;
    const int wave = tid >> 5;
    const int half = lane >> 4;     // K-pair select (WMMA f32 A/B layout)
    const int col  = lane & 15;     // pixel within tile / matrix row m

    long g   = (long)blockIdx.x * 128 + (long)wave * 16;   // global pixel base
    int  b   = (int)(g >> 16);
    int  pix = (int)(g & 65535);

    const float* xw = x + ((long)b << 24) + pix + col + (long)(half * 2) * HW;

    const float* a0p = &W[( 0 + col) * CCH + half * 2];
    const float* a1p = &W[(16 + col) * CCH + half * 2];
    const float* a2p = &W[(32 + col) * CCH + half * 2];
    const float* a3p = &W[(48 + col) * CCH + half * 2];

    v8f acc0 = {}, acc1 = {}, acc2 = {}, acc3 = {};
    float sumsq = 0.f;

#pragma unroll 4
    for (int ks = 0; ks < 64; ++ks) {
        // B frag: 4x16 pixels; lanes 0-15 hold K rows {0,1}, lanes 16-31 {2,3}
        float b0 = __builtin_nontemporal_load(xw);
        float b1 = __builtin_nontemporal_load(xw + HW);
        xw += 4 * HW;
        sumsq = fmaf(b0, b0, sumsq);
        sumsq = fmaf(b1, b1, sumsq);
        v2f bf;  bf.x = b0;  bf.y = b1;

        v2f af0 = *(const v2f*)(a0p + 4 * ks);   // ds_load_2addr_b64 pairs
        v2f af1 = *(const v2f*)(a1p + 4 * ks);
        v2f af2 = *(const v2f*)(a2p + 4 * ks);
        v2f af3 = *(const v2f*)(a3p + 4 * ks);

        acc0 = __builtin_amdgcn_wmma_f32_16x16x4_f32(false, af0, false, bf, (short)0, acc0, false, false);
        acc1 = __builtin_amdgcn_wmma_f32_16x16x4_f32(false, af1, false, bf, (short)0, acc1, false, false);
        acc2 = __builtin_amdgcn_wmma_f32_16x16x4_f32(false, af2, false, bf, (short)0, acc2, false, false);
        acc3 = __builtin_amdgcn_wmma_f32_16x16x4_f32(false, af3, false, bf, (short)0, acc3, false, false);
    }

    // fold ||x||^2 partials across half-waves: lane l and l^16 share a pixel
    float xx = sumsq + __shfl_xor(sumsq, 16);

    // ---- elementwise hyperbolic MLR epilogue + stores ----
    const float eps     = 1e-12f;
    const float sqrtK   = 0.70710678118654752f;        // sqrt(0.5)
    const float maxnorm = (1.0f - 0.001f) / sqrtK;
    const float mn2     = maxnorm * maxnorm;
    float* outw = out + (long)b * NCLS * HW + pix + col;

#pragma unroll
    for (int t = 0; t < 2; ++t) {
        v8f pxv = t ? acc1 : acc0;
        v8f xav = t ? acc3 : acc2;
#pragma unroll
        for (int r = 0; r < 8; ++r) {
            int cls = t * 16 + r + half * 8;  // D row m -> class
            if (cls < NCLS) {
                float px = pxv[r];
                float xa = xav[r] * invS[cls];   // normalize A here
                float pp = ppS[cls];
                float pd = pdS[cls];
                float an = anS[cls];

                float Aq    = 1.0f + 2.0f * KC * px + KC * xx;
                float Bc    = 1.0f - KC * pp;
                float Dq    = fmaxf(1.0f + 2.0f * KC * px + (KC * xx) * (KC * pp), eps);
                float alpha = Aq / Dq;
                float beta  = Bc / Dq;
                float mob   = alpha * alpha * pp + beta * beta * xx
                            + 2.0f * alpha * beta * px;
                float sqm   = sqrtf(mob);
                float proj  = (sqm > maxnorm) ? (maxnorm / fmaxf(sqm, eps)) : 1.0f;
                float mobp  = (sqm < maxnorm) ? mob : mn2;
                float mdot  = (beta * xa + alpha * pd) * proj;
                float lamb  = 2.0f / fmaxf(1.0f - KC * mobp, eps);
                float s     = sqrtK * mdot * lamb;
                float ash   = copysignf(logf(fabsf(s) + sqrtf(fmaf(s, s, 1.0f))), s);
                __builtin_nontemporal_store((2.0f / sqrtK) * an * ash,
                                            outw + (long)cls * HW);
            }
        }
    }
}

extern "C" void kernel_launch(void* const* d_in, const int* in_sizes, int n_in,
                              void* d_out, int out_size, void* d_ws, size_t ws_size,
                              hipStream_t stream) {
    (void)in_sizes; (void)n_in; (void)d_ws; (void)ws_size; (void)out_size;
    const float* x = (const float*)d_in[0];
    const float* P = (const float*)d_in[1];
    const float* A = (const float*)d_in[2];
    float* out = (float*)d_out;
    // 262144 pixels / (8 waves * 16 pixels) = 2048 blocks of 256 threads
    hyper_mlr_fused<<<2048, 256, 0, stream>>>(x, P, A, out);
}